// LightGCN_1288490189549
// MI455X (gfx1250) — compile-verified
//
#include <hip/hip_runtime.h>

// LightGCN propagation for MI455X (gfx1250, wave32).
// Memory-bound SpMM (AI ~0.5 FLOP/B; 76.8MB embedding table is L2-resident).
// Per-row neighbor reduction runs on the WMMA pipe (V_WMMA_F32_16X16X4_F32):
//   A = all-ones (layout-proof), B[k][n] = val_k * x[col_k][n]
//   => D[m][n] = sum_k val_k * x[col_k][n]  for every m  (K-permutation invariant)
// Row bounds are broadcast via readfirstlane so the neighbor loop is a
// *scalar* loop: EXEC is provably all-ones at every WMMA (ISA requirement)
// and there is no exec-mask bookkeeping in the hot loop.

#define EMB 64
#define LAYERS 3

typedef __attribute__((ext_vector_type(2))) float v2f;
typedef __attribute__((ext_vector_type(8))) float v8f;

// ---------------------------------------------------------------------------
// Kernel 1: CSR row_ptr from sorted COO rows via per-row lower_bound.
// ---------------------------------------------------------------------------
__global__ void build_row_ptr(const int* __restrict__ rows, int nnz, int n,
                              int* __restrict__ row_ptr) {
    int r = blockIdx.x * blockDim.x + threadIdx.x;
    if (r > n) return;
    int lo = 0, hi = nnz;
    while (lo < hi) {
        int mid = (lo + hi) >> 1;
        if (rows[mid] < r) lo = mid + 1; else hi = mid;
    }
    row_ptr[r] = lo;
}

// ---------------------------------------------------------------------------
// WMMA SpMM: one wave32 per output row. 4 neighbors per K-step, 4 dim-chunks
// of 16 per step (64-wide embedding). Lane L: half = L>>4, n16 = L&15.
// A is all-ones so D = column-sums of B regardless of the exact K layout.
// SRC_CONCAT: read layer-0 x from the two input tables.
// FINAL: fuse out = (x0 + x1 + x2 + y3) * scale.
// ---------------------------------------------------------------------------
template <int SRC_CONCAT, int FINAL>
__global__ void spmm_wmma(const float* __restrict__ src,   // gather source (!SRC_CONCAT)
                          const float* __restrict__ ue,
                          const float* __restrict__ ie,
                          const float* __restrict__ xA,    // x1 (FINAL only)
                          const float* __restrict__ xB,    // x2 (FINAL only)
                          const int* __restrict__ row_ptr,
                          const int* __restrict__ cols,
                          const float* __restrict__ vals,
                          float* __restrict__ dst,
                          int n, int n_users, float scale) {
    // Row id is wave-uniform; force the compiler to see it that way.
    int wave = __builtin_amdgcn_readfirstlane(
        (int)((blockIdx.x * blockDim.x + threadIdx.x) >> 5));
    if (wave >= n) return;                    // scalar branch, EXEC untouched
    int lane = threadIdx.x & 31;
    int half = lane >> 4;                     // 0: K rows {0,1}, 1: K rows {2,3}
    int n16  = lane & 15;                     // N index within chunk
    int ka = half * 2, kb = ka + 1;

    // Scalar (SGPR) loop bounds -> uniform loop, EXEC all-ones at every WMMA.
    int beg = __builtin_amdgcn_readfirstlane(row_ptr[wave]);
    int end = __builtin_amdgcn_readfirstlane(row_ptr[wave + 1]);

    v8f acc0 = {}, acc1 = {}, acc2 = {}, acc3 = {};
    v2f aone; aone.x = 1.0f; aone.y = 1.0f;   // all-ones A matrix

    for (int i = beg; i < end; i += 4) {      // scalar loop (s_cmp/s_cbranch)
        // Branchless tail padding: clamped in-bounds indices, unconditional
        // loads, then cndmask the val to zero for padded slots.
        bool oka = (i + ka) < end;
        bool okb = (i + kb) < end;
        int  ja  = oka ? (i + ka) : beg;
        int  jb  = okb ? (i + kb) : beg;
        int  ca  = cols[ja];
        int  cb  = cols[jb];
        float va = vals[ja];
        float vb = vals[jb];
        va = oka ? va : 0.0f;                 // v_cndmask, no predicated load
        vb = okb ? vb : 0.0f;

        const float* xa;
        const float* xb;
        if (SRC_CONCAT) {
            xa = (ca < n_users) ? ue + (size_t)ca * EMB
                                : ie + (size_t)(ca - n_users) * EMB;
            xb = (cb < n_users) ? ue + (size_t)cb * EMB
                                : ie + (size_t)(cb - n_users) * EMB;
        } else {
            xa = src + (size_t)ca * EMB;
            xb = src + (size_t)cb * EMB;
        }

        // Prefetch next neighbor group's rows (global_prefetch_b8);
        // guard is scalar now.
        if (i + 4 < end) {
            int jn = i + 4 + ka;
            jn = jn < end ? jn : end - 1;
            int cn = cols[jn];
            const float* xn;
            if (SRC_CONCAT)
                xn = (cn < n_users) ? ue + (size_t)cn * EMB
                                    : ie + (size_t)(cn - n_users) * EMB;
            else
                xn = src + (size_t)cn * EMB;
            __builtin_prefetch(xn + n16, 0, 1);
        }

        v2f b;
        b.x = va * xa[n16];        b.y = vb * xb[n16];
        acc0 = __builtin_amdgcn_wmma_f32_16x16x4_f32(false, aone, false, b,
                                                     (short)0, acc0, false, false);
        b.x = va * xa[16 + n16];   b.y = vb * xb[16 + n16];
        acc1 = __builtin_amdgcn_wmma_f32_16x16x4_f32(false, aone, false, b,
                                                     (short)0, acc1, false, false);
        b.x = va * xa[32 + n16];   b.y = vb * xb[32 + n16];
        acc2 = __builtin_amdgcn_wmma_f32_16x16x4_f32(false, aone, false, b,
                                                     (short)0, acc2, false, false);
        b.x = va * xa[48 + n16];   b.y = vb * xb[48 + n16];
        acc3 = __builtin_amdgcn_wmma_f32_16x16x4_f32(false, aone, false, b,
                                                     (short)0, acc3, false, false);
    }

    // All D rows identical; acc[0] in lane L is the chunk value at N = L&15.
    // Lanes 0-15 cover chunks {0,1}, lanes 16-31 chunks {2,3}: each store
    // writes two coalesced 64B segments.
    int d0 = half * 32 + n16;   // dim 0..15  or 32..47
    int d1 = d0 + 16;           // dim 16..31 or 48..63
    float y0 = half ? acc2[0] : acc0[0];
    float y1 = half ? acc3[0] : acc1[0];

    size_t base = (size_t)wave * EMB;
    if (FINAL) {
        const float* x0 = (wave < n_users) ? ue + (size_t)wave * EMB
                                           : ie + (size_t)(wave - n_users) * EMB;
        dst[base + d0] = (x0[d0] + xA[base + d0] + xB[base + d0] + y0) * scale;
        dst[base + d1] = (x0[d1] + xA[base + d1] + xB[base + d1] + y1) * scale;
    } else {
        dst[base + d0] = y0;
        dst[base + d1] = y1;
    }
}

// ---------------------------------------------------------------------------
extern "C" void kernel_launch(void* const* d_in, const int* in_sizes, int n_in,
                              void* d_out, int out_size, void* d_ws, size_t ws_size,
                              hipStream_t stream) {
    const float* ue   = (const float*)d_in[0];  // user_emb   [N_USERS, 64]
    const float* ie   = (const float*)d_in[1];  // item_emb   [N_ITEMS, 64]
    const int*   rows = (const int*)d_in[2];    // adj_rows   [NNZ] (sorted)
    const int*   cols = (const int*)d_in[3];    // adj_cols   [NNZ]
    const float* vals = (const float*)d_in[4];  // adj_vals   [NNZ]

    const int n_users = in_sizes[0] / EMB;
    const int n_items = in_sizes[1] / EMB;
    const int n       = n_users + n_items;
    const int nnz     = in_sizes[2];
    float* out = (float*)d_out;

    // Workspace: row_ptr | x1 | x2   (~1.2MB + 2 * 76.8MB)
    char* ws = (char*)d_ws;
    size_t rp_bytes = (((size_t)(n + 1) * sizeof(int)) + 255) & ~(size_t)255;
    int*   row_ptr = (int*)ws;
    float* xA = (float*)(ws + rp_bytes);
    float* xB = xA + (size_t)n * EMB;

    // 1) CSR pointers (sorted rows -> binary search, no atomics).
    build_row_ptr<<<(n + 1 + 255) / 256, 256, 0, stream>>>(rows, nnz, n, row_ptr);

    // 2) Three propagation layers; one wave32 per row, 8 rows per block.
    const int threads = 256;
    const int rows_per_block = threads / 32;
    const int blocks = (n + rows_per_block - 1) / rows_per_block;
    const float scale = 1.0f / (float)(LAYERS + 1);

    // layer 1: x1 = A . x0   (x0 read directly from input tables)
    spmm_wmma<1, 0><<<blocks, threads, 0, stream>>>(nullptr, ue, ie, nullptr, nullptr,
                                                    row_ptr, cols, vals, xA,
                                                    n, n_users, scale);
    // layer 2: x2 = A . x1
    spmm_wmma<0, 0><<<blocks, threads, 0, stream>>>(xA, ue, ie, nullptr, nullptr,
                                                    row_ptr, cols, vals, xB,
                                                    n, n_users, scale);
    // layer 3 (fused): out = (x0 + x1 + x2 + A . x2) / (LAYERS+1)
    spmm_wmma<0, 1><<<blocks, threads, 0, stream>>>(xB, ue, ie, xA, xB,
                                                    row_ptr, cols, vals, out,
                                                    n, n_users, scale);
}